// HistogramNd_46411416600801
// MI455X (gfx1250) — compile-verified
//
#include <hip/hip_runtime.h>
#include <hip/hip_bf16.h>

// 3D weighted histogram for MI455X (gfx1250, wave32).
// d_in: values[N*3] f32, weights[N] f32, hist_sum[BINS] f32, oob_sum[1] f32,
//       range_min[3] f32, range_max[3] f32, n_bins[3] i32, stride[3] i32
// d_out: [BINS] histogram followed by [1] oob scalar (BINS+1 floats).

typedef __attribute__((ext_vector_type(4))) float v4f;

__global__ void hist_init_kernel(const float* __restrict__ hist_in,
                                 const float* __restrict__ oob_in,
                                 float* __restrict__ out, int bins) {
  int i = blockIdx.x * blockDim.x + threadIdx.x;
  if (i < bins) out[i] = hist_in[i];
  if (i == bins) out[bins] = oob_in[0];
}

__global__ __launch_bounds__(256) void hist_accum_kernel(
    const float* __restrict__ values,
    const float* __restrict__ weights,
    const float* __restrict__ range_min,
    const float* __restrict__ range_max,
    const int*   __restrict__ n_bins,
    const int*   __restrict__ stride,
    float* __restrict__ out,
    long N, int bins) {
  // Uniform per-kernel parameters (tiny, L2-resident loads).
  const float lo0 = range_min[0], lo1 = range_min[1], lo2 = range_min[2];
  const float inv0 = 1.0f / (range_max[0] - lo0);
  const float inv1 = 1.0f / (range_max[1] - lo1);
  const float inv2 = 1.0f / (range_max[2] - lo2);
  const int nb0 = n_bins[0], nb1 = n_bins[1], nb2 = n_bins[2];
  const float nb0f = (float)nb0, nb1f = (float)nb1, nb2f = (float)nb2;
  const int st0 = stride[0], st1 = stride[1], st2 = stride[2];

  float oob = 0.0f;

  auto do_point = [&](float x, float y, float z, float wt) {
    // Match reference: floor(((v - lo) / (hi - lo)) * n_bins) -> int32
    int b0 = (int)floorf(((x - lo0) * inv0) * nb0f);
    int b1 = (int)floorf(((y - lo1) * inv1) * nb1f);
    int b2 = (int)floorf(((z - lo2) * inv2) * nb2f);
    bool ok = ((unsigned)b0 < (unsigned)nb0) &
              ((unsigned)b1 < (unsigned)nb1) &
              ((unsigned)b2 < (unsigned)nb2);
    if (ok) {
      int flat = b0 * st0 + b1 * st1 + b2 * st2;
      // No-return agent-scope atomic -> global_atomic_add_f32 (STOREcnt path)
      (void)__hip_atomic_fetch_add(&out[flat], wt, __ATOMIC_RELAXED,
                                   __HIP_MEMORY_SCOPE_AGENT);
    } else {
      oob += wt;
    }
  };

  const v4f* __restrict__ w4 = (const v4f*)weights;
  const v4f* __restrict__ v4 = (const v4f*)values;

  const long gid = (long)blockIdx.x * blockDim.x + threadIdx.x;
  const long gsz = (long)gridDim.x * blockDim.x;
  const long nQuad = N >> 2;  // groups of 4 points (48B values + 16B weights)

  for (long q = gid; q < nQuad; q += gsz) {
    // Prefetch one grid-stride ahead (global_prefetch_b8, speculative).
    long qn = q + gsz;
    if (qn < nQuad) {
      __builtin_prefetch(&w4[qn], 0, 1);
      __builtin_prefetch(&v4[qn * 3 + 0], 0, 1);
      __builtin_prefetch(&v4[qn * 3 + 2], 0, 1);
    }
    // Streaming 128-bit non-temporal loads: don't pollute L2 — keep the
    // 1 MB histogram resident there for the atomics.
    v4f w = __builtin_nontemporal_load(&w4[q]);
    v4f a = __builtin_nontemporal_load(&v4[q * 3 + 0]);
    v4f b = __builtin_nontemporal_load(&v4[q * 3 + 1]);
    v4f c = __builtin_nontemporal_load(&v4[q * 3 + 2]);

    do_point(a.x, a.y, a.z, w.x);
    do_point(a.w, b.x, b.y, w.y);
    do_point(b.z, b.w, c.x, w.z);
    do_point(c.y, c.z, c.w, w.w);
  }

  // Scalar tail (N not a multiple of 4).
  for (long i = (nQuad << 2) + gid; i < N; i += gsz) {
    float x = values[i * 3 + 0];
    float y = values[i * 3 + 1];
    float z = values[i * 3 + 2];
    do_point(x, y, z, weights[i]);
  }

  // OOB: wave32 shuffle reduction -> LDS block reduction -> one conditional
  // atomic per block (skipped when the block saw no out-of-range points).
  for (int off = 16; off > 0; off >>= 1)
    oob += __shfl_xor(oob, off, 32);

  __shared__ float sred[8];  // 256 threads = 8 wave32s
  const int lane = threadIdx.x & 31;
  const int wid  = threadIdx.x >> 5;
  if (lane == 0) sred[wid] = oob;
  __syncthreads();
  if (wid == 0) {
    int nwaves = (int)(blockDim.x >> 5);
    float v = (lane < nwaves) ? sred[lane] : 0.0f;
    for (int off = 4; off > 0; off >>= 1)
      v += __shfl_xor(v, off, 32);
    if (lane == 0 && v != 0.0f) {
      (void)__hip_atomic_fetch_add(&out[bins], v, __ATOMIC_RELAXED,
                                   __HIP_MEMORY_SCOPE_AGENT);
    }
  }
}

extern "C" void kernel_launch(void* const* d_in, const int* in_sizes, int n_in,
                              void* d_out, int out_size, void* d_ws, size_t ws_size,
                              hipStream_t stream) {
  const float* values  = (const float*)d_in[0];
  const float* weights = (const float*)d_in[1];
  const float* hist_in = (const float*)d_in[2];
  const float* oob_in  = (const float*)d_in[3];
  const float* rmin    = (const float*)d_in[4];
  const float* rmax    = (const float*)d_in[5];
  const int*   nbins   = (const int*)d_in[6];
  const int*   stridep = (const int*)d_in[7];

  const long N    = (long)in_sizes[1];  // number of points (weights count)
  const int  bins = in_sizes[2];        // total histogram bins

  float* out = (float*)d_out;

  // 1) Initialize d_out = [hist_sum_in, oob_sum_in] (d_out is poisoned).
  int initThreads = 256;
  int initBlocks  = (bins + 1 + initThreads - 1) / initThreads;
  hipLaunchKernelGGL(hist_init_kernel, dim3(initBlocks), dim3(initThreads), 0,
                     stream, hist_in, oob_in, out, bins);

  // 2) Scatter-add accumulate. 1024 blocks x 256 threads = 8192 wave32s,
  //    ~8 quad-iterations per thread: enough concurrency to saturate the
  //    23.3 TB/s read stream while keeping the prefetch-ahead loop live.
  hipLaunchKernelGGL(hist_accum_kernel, dim3(1024), dim3(256), 0, stream,
                     values, weights, rmin, rmax, nbins, stridep, out, N, bins);
}